// GCN_90984587198652
// MI455X (gfx1250) — compile-verified
//
#include <hip/hip_runtime.h>

// ---------------------------------------------------------------------------
// GCN forward for MI455X (gfx1250): f32 WMMA GEMMs + L2-resident atomic SpMM
// ---------------------------------------------------------------------------

typedef __attribute__((ext_vector_type(2))) float v2f;
typedef __attribute__((ext_vector_type(8))) float v8f;

#define N_NODES  100000
#define N_EDGES  1600000
#define N_INPUT  256
#define N_HIDDEN 128
#define N_OUTPUT 64

// ---------------------------------------------------------------------------
// Dense GEMM: C[16-row slab] = A[m0:m0+16, 0:K] * B[K, NOUT]
// One workgroup per 16-row slab; one wave32 per 16-col tile; K stepped by 4
// with V_WMMA_F32_16X16X4_F32. A slab staged in LDS with +4 dword pad.
// ---------------------------------------------------------------------------
template <int K, int NOUT>
__global__ __launch_bounds__(32 * (NOUT / 16))
void gcn_gemm_wmma_f32(const float* __restrict__ A,
                       const float* __restrict__ B,
                       float* __restrict__ C) {
    constexpr int WAVES = NOUT / 16;
    constexpr int NTHR  = 32 * WAVES;
    constexpr int LDA   = K + 4;          // pad: K%64==0 would 16-way conflict
    __shared__ float tile[16 * LDA];

    const int m0  = blockIdx.x * 16;
    const int tid = threadIdx.x;

    // Cooperative, coalesced float4 staging of the 16 x K A-slab.
    constexpr int NV4 = 16 * (K / 4);
    #pragma unroll
    for (int i4 = tid; i4 < NV4; i4 += NTHR) {
        const int row = i4 / (K / 4);
        const int c4  = (i4 % (K / 4)) * 4;
        const float4 v = *reinterpret_cast<const float4*>(
            A + (size_t)(m0 + row) * K + c4);
        *reinterpret_cast<float4*>(&tile[row * LDA + c4]) = v;
    }
    __syncthreads();

    const int lane  = tid & 31;
    const int wave  = tid >> 5;
    const int n0    = wave * 16;
    const int fr    = lane & 15;          // M for A-frag, N for B-frag
    const int khalf = (lane >> 4) * 2;    // lanes 16-31 hold K+2,K+3

    v8f acc = {};
    #pragma unroll 4
    for (int k = 0; k < K; k += 4) {
        v2f a, b;
        // A(16x4): lane fr = row M, VGPR j -> K = k + khalf + j (from LDS)
        a.x = tile[fr * LDA + k + khalf];
        a.y = tile[fr * LDA + k + khalf + 1];
        // B(4x16): lane fr = col N, VGPR j -> K = k + khalf + j (from L2-hot B)
        const float* bp = B + (size_t)(k + khalf) * NOUT + n0 + fr;
        b.x = bp[0];
        b.y = bp[NOUT];
        acc = __builtin_amdgcn_wmma_f32_16x16x4_f32(
            /*neg_a=*/false, a, /*neg_b=*/false, b,
            /*c_mod=*/(short)0, acc, /*reuse_a=*/false, /*reuse_b=*/false);
    }

    // C/D layout: VGPR v -> M = v + 8*(lane>>4), N = fr
    const int mhi = (lane >> 4) * 8;
    float* cp = C + (size_t)(m0 + mhi) * NOUT + n0 + fr;
    #pragma unroll
    for (int v = 0; v < 8; ++v)
        cp[(size_t)v * NOUT] = acc[v];
}

// ---------------------------------------------------------------------------
// SpMM (COO scatter-add): out[r, :] += w * H[c, :]  over edges.
// One thread per (edge, 4-feature chunk): float4 gather (L2-resident H),
// 4x native GLOBAL_ATOMIC_ADD_F32 into the accumulator (resolved at L2).
// unsafeAtomicAdd forces the hardware fp atomic (no CAS retry loop).
// ---------------------------------------------------------------------------
template <int F>
__global__ __launch_bounds__(256)
void gcn_spmm_atomic(const int* __restrict__ rows,
                     const int* __restrict__ cols,
                     const float* __restrict__ wts,
                     const float* __restrict__ H,
                     float* __restrict__ out) {
    constexpr int CH = F / 4;             // float4 chunks per row (power of 2)
    const long long t = (long long)blockIdx.x * blockDim.x + threadIdx.x;
    const long long total = (long long)N_EDGES * CH;
    if (t >= total) return;
    const int e = (int)(t >> (CH == 32 ? 5 : 4));
    const int f = ((int)t & (CH - 1)) * 4;

    const int   r  = rows[e];
    const int   c  = cols[e];
    const float ew = wts[e];

    const float4 h = *reinterpret_cast<const float4*>(H + (size_t)c * F + f);
    float* o = out + (size_t)r * F + f;
    unsafeAtomicAdd(o + 0, ew * h.x);
    unsafeAtomicAdd(o + 1, ew * h.y);
    unsafeAtomicAdd(o + 2, ew * h.z);
    unsafeAtomicAdd(o + 3, ew * h.w);
}

// ---------------------------------------------------------------------------
// Zero an accumulator buffer (count must be a multiple of 4 floats).
// ---------------------------------------------------------------------------
__global__ __launch_bounds__(256)
void gcn_zero_f32(float* __restrict__ p, int n4) {
    const int t = blockIdx.x * blockDim.x + threadIdx.x;
    if (t < n4) {
        float4 z; z.x = 0.f; z.y = 0.f; z.z = 0.f; z.w = 0.f;
        reinterpret_cast<float4*>(p)[t] = z;
    }
}

// ---------------------------------------------------------------------------
extern "C" void kernel_launch(void* const* d_in, const int* in_sizes, int n_in,
                              void* d_out, int out_size, void* d_ws, size_t ws_size,
                              hipStream_t stream) {
    const float* X  = (const float*)d_in[0];   // [N_NODES, 256]
    const int*   er = (const int*)  d_in[1];   // [N_EDGES]
    const int*   ec = (const int*)  d_in[2];   // [N_EDGES]
    const float* ew = (const float*)d_in[3];   // [N_EDGES]
    const float* W1 = (const float*)d_in[4];   // [256, 128]
    const float* W2 = (const float*)d_in[5];   // [128, 64]
    float* out = (float*)d_out;                // [N_NODES, 64]

    char* ws = (char*)d_ws;
    const size_t slab = (size_t)N_NODES * N_HIDDEN * sizeof(float); // 51.2 MB
    float* H1 = (float*)(ws);          // X @ W1          [N, 128]
    float* S1 = (float*)(ws + slab);   // A_hat @ H1      [N, 128]
    float* H2 = (float*)(ws);          // S1 @ W2 (reuses H1 region) [N, 64]

    constexpr int MB = N_NODES / 16;   // 6250 row-slabs, exact

    // 1) H1 = X @ W1                                  (WMMA f32, 8 waves/blk)
    gcn_gemm_wmma_f32<N_INPUT, N_HIDDEN><<<MB, 256, 0, stream>>>(X, W1, H1);

    // 2) S1 = 0 ; S1[r] += w * H1[c]
    {
        const int n4 = N_NODES * N_HIDDEN / 4;         // 3.2M float4
        gcn_zero_f32<<<(n4 + 255) / 256, 256, 0, stream>>>(S1, n4);
        const long long thr = (long long)N_EDGES * (N_HIDDEN / 4);
        gcn_spmm_atomic<N_HIDDEN>
            <<<(int)((thr + 255) / 256), 256, 0, stream>>>(er, ec, ew, H1, S1);
    }

    // 3) H2 = S1 @ W2                                 (WMMA f32, 4 waves/blk)
    gcn_gemm_wmma_f32<N_HIDDEN, N_OUTPUT><<<MB, 128, 0, stream>>>(S1, W2, H2);

    // 4) out = 0 ; out[r] += w * H2[c]
    {
        const int n4 = N_NODES * N_OUTPUT / 4;         // 1.6M float4
        gcn_zero_f32<<<(n4 + 255) / 256, 256, 0, stream>>>(out, n4);
        const long long thr = (long long)N_EDGES * (N_OUTPUT / 4);
        gcn_spmm_atomic<N_OUTPUT>
            <<<(int)((thr + 255) / 256), 256, 0, stream>>>(er, ec, ew, H2, out);
    }
}